// PixelWiseRNNEfficient_4612794875905
// MI455X (gfx1250) — compile-verified
//
#include <hip/hip_runtime.h>

// Per-pixel diagonal RNN over T=50 timesteps.
// x: (1, 50, 8, 512, 512) f32; W_ih/W_hh/b_hh: (8, 512, 512) f32.
// Purely elementwise -> bandwidth-bound (~824 MB streamed, ~35 us floor
// at 23.3 TB/s). Strategy: 1 thread owns 4 consecutive pixels, weights
// held in VGPRs, recurrence state lives in registers across the T-loop,
// 128-bit coalesced non-temporal loads/stores, global_prefetch for the
// serial time dimension.

typedef float v4f __attribute__((ext_vector_type(4)));

namespace {
constexpr int   kT      = 50;
constexpr int   kN      = 8 * 512 * 512;   // pixels per timestep = 2,097,152
constexpr int   kVec    = 4;               // pixels per thread (float4)
constexpr int   kBlock  = 256;             // 8 wave32s per block
constexpr float kSlope  = 0.01f;
}

__global__ __launch_bounds__(kBlock) void
pixel_diag_rnn_kernel(const float* __restrict__ x,
                      const float* __restrict__ w_ih,
                      const float* __restrict__ w_hh,
                      const float* __restrict__ b_hh,
                      float* __restrict__ out)
{
    const long long p = (long long)(blockIdx.x * kBlock + threadIdx.x) * kVec;
    if (p >= kN) return;

    // Per-pixel weights: loaded exactly once, pinned in VGPRs for all 50 steps.
    const v4f wi = __builtin_nontemporal_load((const v4f*)(w_ih + p));
    const v4f wh = __builtin_nontemporal_load((const v4f*)(w_hh + p));
    const v4f bb = __builtin_nontemporal_load((const v4f*)(b_hh + p));

    v4f h = {0.f, 0.f, 0.f, 0.f};

    const float* __restrict__ xp = x   + p;
    float*       __restrict__ op = out + p;

#pragma unroll 2
    for (int t = 0; t < kT; ++t) {
        // Cover HBM latency on the serial time axis: pull x[t+2] toward L2.
        // Lowers to gfx1250 global_prefetch_b8 (no LOADcnt cost).
        if (t + 2 < kT) {
            __builtin_prefetch(xp + (long long)(t + 2) * kN, /*rw=*/0, /*loc=*/1);
        }

        // Streaming, single-use data -> non-temporal (TH=NT) 128-bit load.
        const v4f xv =
            __builtin_nontemporal_load((const v4f*)(xp + (long long)t * kN));

        v4f hn;
#pragma unroll
        for (int i = 0; i < kVec; ++i) {
            float v = fmaf(wh[i], h[i], bb[i]);   // W_hh*h + b
            v       = fmaf(wi[i], xv[i], v);      // + W_ih*x
            // leaky_relu with slope<1: max(v, slope*v)
            hn[i]   = fmaxf(v, kSlope * v);
        }
        h = hn;

        // Output is write-once, never re-read -> non-temporal 128-bit store.
        __builtin_nontemporal_store(h, (v4f*)(op + (long long)t * kN));
    }
}

extern "C" void kernel_launch(void* const* d_in, const int* in_sizes, int n_in,
                              void* d_out, int out_size, void* d_ws, size_t ws_size,
                              hipStream_t stream) {
    (void)in_sizes; (void)n_in; (void)d_ws; (void)ws_size; (void)out_size;

    const float* x    = (const float*)d_in[0];
    const float* w_ih = (const float*)d_in[1];
    const float* w_hh = (const float*)d_in[2];
    const float* b_hh = (const float*)d_in[3];
    float*       out  = (float*)d_out;

    const int threads = kN / kVec;                 // 524,288
    const int blocks  = threads / kBlock;          // 2,048
    pixel_diag_rnn_kernel<<<blocks, kBlock, 0, stream>>>(x, w_ih, w_hh, b_hh, out);
}